// TrvTimesCorrection_19490561589306
// MI455X (gfx1250) — compile-verified
//
#include <hip/hip_runtime.h>
#include <math.h>

typedef __attribute__((ext_vector_type(2))) float v2f;
typedef __attribute__((ext_vector_type(8))) float v8f;
typedef __attribute__((ext_vector_type(4))) int   v4i;

typedef __attribute__((address_space(1))) v4i* as1_v4i_p;  // global-AS int4*
typedef __attribute__((address_space(3))) v4i* as3_v4i_p;  // LDS-AS int4*

#define K_NN   15
#define N_STA  200
#define N_SRC  4000
#define N_GRID 10000
#define N_REF  500

#if defined(__gfx1250__) && __has_builtin(__builtin_amdgcn_global_load_async_to_lds_b128)
#define HAVE_ASYNC_LDS 1
#else
#define HAVE_ASYNC_LDS 0
#endif

__device__ __forceinline__ void wait_async_zero() {
#if __has_builtin(__builtin_amdgcn_s_wait_asynccnt)
  __builtin_amdgcn_s_wait_asynccnt(0);
#else
  asm volatile("s_wait_asynccnt 0" ::: "memory");
#endif
}

// ---------------------------------------------------------------------------
// Kernel 1: nearest reference location per station (200 x 500 argmin)
// ---------------------------------------------------------------------------
__global__ void sta_ind_kernel(const float* __restrict__ sta,
                               const float* __restrict__ locs_ref,
                               int* __restrict__ sta_ind) {
  int n = blockIdx.x * blockDim.x + threadIdx.x;
  if (n >= N_STA) return;
  float sx = sta[n * 3 + 0] * 1e-3f;
  float sy = sta[n * 3 + 1] * 1e-3f;
  float sz = sta[n * 3 + 2] * 1e-3f;
  float best = 3.4e38f;
  int bidx = 0;
  for (int r = 0; r < N_REF; ++r) {
    float dx = sx - locs_ref[r * 3 + 0] * 1e-3f;
    float dy = sy - locs_ref[r * 3 + 1] * 1e-3f;
    float dz = sz - locs_ref[r * 3 + 2] * 1e-3f;
    float d = dx * dx + dy * dy + dz * dz;
    if (d < best) { best = d; bidx = r; }
  }
  sta_ind[n] = bidx;
}

// ---------------------------------------------------------------------------
// Kernel 2: k-NN of each source among grid nodes.
//
// Per wave: 16 sources. WMMA roles: A = grid chunk (16 nodes x K=4, rebuilt
// each iteration from LDS-staged grid), B = source tile (K=4 x 16 sources,
// loop-invariant registers). C/D layout: lane l holds N = l%16 (its source)
// and rows M = v + 8*(l>=16), v=0..7 -> each lane keeps a private sorted
// top-15 over its half of the grid; lane n / lane n+16 lists merge at the end.
// Inner loop: barrier-free; grid staged once via async Global->LDS.
// ---------------------------------------------------------------------------
__global__ __launch_bounds__(256) void knn_kernel(const float* __restrict__ src,
                                                  const float* __restrict__ x_grid,
                                                  int* __restrict__ nn_idx) {
  __shared__ __align__(16) float grid_lds[N_GRID * 3];  // 120 KB (raw meters)
  __shared__ float gn2_lds[N_GRID];                     //  40 KB (km^2 norms)

  const int lane = threadIdx.x & 31;
  const int wave = threadIdx.x >> 5;
  const int srcBase = (blockIdx.x * 8 + wave) * 16;
  const bool active = (srcBase < N_SRC);

  // ---- stage x_grid into LDS (async Global->LDS on gfx1250) --------------
#if HAVE_ASYNC_LDS
  for (int i4 = threadIdx.x; i4 < (N_GRID * 3) / 4; i4 += blockDim.x) {
    __builtin_amdgcn_global_load_async_to_lds_b128(
        (as1_v4i_p)(x_grid + i4 * 4),
        (as3_v4i_p)(grid_lds + i4 * 4),
        0, 0);
  }
  wait_async_zero();
#else
  {
    const float4* gs = (const float4*)x_grid;
    float4* ls = (float4*)grid_lds;
    for (int i4 = threadIdx.x; i4 < (N_GRID * 3) / 4; i4 += blockDim.x)
      ls[i4] = gs[i4];
  }
#endif
  __syncthreads();

  // ---- precompute grid-node squared norms (km^2) -------------------------
  for (int g = threadIdx.x; g < N_GRID; g += blockDim.x) {
    float x = grid_lds[g * 3 + 0] * 1e-3f;
    float y = grid_lds[g * 3 + 1] * 1e-3f;
    float z = grid_lds[g * 3 + 2] * 1e-3f;
    gn2_lds[g] = x * x + y * y + z * z;
  }
  __syncthreads();

  const int m    = lane & 15;    // source column N owned by this lane
  const int half = lane >> 4;    // 0/1 -> K pair {0,1} / {2,3}; rows 0-7 / 8-15

  // ---- B tile (sources), loop-invariant; plus this lane's |src|^2 --------
  v2f bsrc = {0.0f, 0.0f};
  float sn2 = 0.0f;
  if (active) {
    float s0 = src[(srcBase + m) * 3 + 0] * 1e-3f;
    float s1 = src[(srcBase + m) * 3 + 1] * 1e-3f;
    float s2 = src[(srcBase + m) * 3 + 2] * 1e-3f;
    bsrc.x = half ? s2 : s0;
    bsrc.y = half ? 0.0f : s1;
    sn2 = s0 * s0 + s1 * s1 + s2 * s2;
  }

  // ---- per-lane sorted top-15 over this lane's half of the grid ----------
  float bd[K_NN];
  int   bi[K_NN];
#pragma unroll
  for (int t = 0; t < K_NN; ++t) { bd[t] = 3.4e38f; bi[t] = 0; }

  if (active) {
    for (int c = 0; c < N_GRID / 16; ++c) {
      // A tile: 16 grid nodes x K4 (K=3 zero-padded), from LDS
      const int node = c * 16 + m;
      float g0 = grid_lds[node * 3 + 0] * 1e-3f;
      float g1 = grid_lds[node * 3 + 1] * 1e-3f;
      float g2 = grid_lds[node * 3 + 2] * 1e-3f;
      v2f a;
      a.x = half ? g2 : g0;
      a.y = half ? 0.0f : g1;

      v8f acc = {};
      acc = __builtin_amdgcn_wmma_f32_16x16x4_f32(
          /*neg_a=*/false, a, /*neg_b=*/false, bsrc,
          /*c_mod=*/(short)0, acc, /*reuse_a=*/false, /*reuse_b=*/false);

      const int gbase = c * 16 + 8 * half;   // rows M this lane holds
#pragma unroll
      for (int v = 0; v < 8; ++v) {
        float d2 = gn2_lds[gbase + v] + sn2 - 2.0f * acc[v];
        int g = gbase + v;
        if (d2 < bd[K_NN - 1]) {             // replace worst, one bubble pass
          bd[K_NN - 1] = d2; bi[K_NN - 1] = g;
#pragma unroll
          for (int t = K_NN - 1; t >= 1; --t) {
            if (bd[t] < bd[t - 1]) {
              float td = bd[t]; bd[t] = bd[t - 1]; bd[t - 1] = td;
              int   ti = bi[t]; bi[t] = bi[t - 1]; bi[t - 1] = ti;
            }
          }
        }
      }
    }
  }

  // ---- merge lane n and lane n+16 lists (reuse grid_lds as scratch) ------
  __syncthreads();   // everyone past the loop; grid_lds free for reuse

  float* mdist = grid_lds + wave * 960;                // 32*15 dists
  int*   midx  = (int*)(grid_lds + wave * 960 + 480);  // 32*15 indices
  if (active) {
#pragma unroll
    for (int t = 0; t < K_NN; ++t) {
      mdist[lane * K_NN + t] = bd[t];
      midx [lane * K_NN + t] = bi[t];
    }
  }
  __syncthreads();

  if (active && lane < 16) {
    const float* dA = mdist + m * K_NN;
    const float* dB = mdist + (m + 16) * K_NN;
    const int*   iA = midx + m * K_NN;
    const int*   iB = midx + (m + 16) * K_NN;
    int i = 0, j = 0;
    for (int t = 0; t < K_NN; ++t) {
      float da = dA[i], db = dB[j];
      bool takeA = (da <= db);
      int gsel = takeA ? iA[i] : iB[j];
      i += takeA; j += !takeA;
      nn_idx[(srcBase + m) * K_NN + t] = gsel;
    }
  }
}

// ---------------------------------------------------------------------------
// Kernel 3: edge weights + normalized scatter-sum + travel time.
// One thread per (source, station); 15 gathered edges each (L2-resident data).
// ---------------------------------------------------------------------------
__device__ __forceinline__ float softplus_f(float x) {
  return (x > 20.0f) ? x : log1pf(__expf(x));
}

__global__ void corr_kernel(const float* __restrict__ sta,
                            const float* __restrict__ src,
                            const float* __restrict__ x_grid,
                            const float* __restrict__ coefs,
                            const float* __restrict__ coefs_ker,
                            const int* __restrict__ sta_ind,
                            const int* __restrict__ nn_idx,
                            float* __restrict__ out) {
  int idx = blockIdx.x * blockDim.x + threadIdx.x;
  if (idx >= N_SRC * N_STA) return;
  int s = idx / N_STA;
  int n = idx - s * N_STA;

  float px = src[s * 3 + 0] * 1e-3f;
  float py = src[s * 3 + 1] * 1e-3f;
  float pz = src[s * 3 + 2] * 1e-3f;
  float dxs = px - sta[n * 3 + 0] * 1e-3f;
  float dys = py - sta[n * 3 + 1] * 1e-3f;
  float dzs = pz - sta[n * 3 + 2] * 1e-3f;
  float dist = sqrtf(dxs * dxs + dys * dys + dzs * dzs + 1e-12f);

  int r = sta_ind[n];
  float wsum = 0.0f, c0 = 0.0f, c1 = 0.0f;
#pragma unroll
  for (int e = 0; e < K_NN; ++e) {
    int g = nn_idx[s * K_NN + e];
    float ddx = x_grid[g * 3 + 0] * 1e-3f - px;
    float ddy = x_grid[g * 3 + 1] * 1e-3f - py;
    float ddz = x_grid[g * 3 + 2] * 1e-3f - pz;
    const float* kp = coefs_ker + (size_t)g * (N_REF * 3) + r * 3;
    float k0 = softplus_f(kp[0]);
    float k1 = softplus_f(kp[1]);
    float k2 = softplus_f(kp[2]);
    float t0 = ddx / k0, t1 = ddy / k1, t2 = ddz / k2;
    float w = __expf(-0.5f * (t0 * t0 + t1 * t1 + t2 * t2));
    wsum += w;
    const float* cp = coefs + (size_t)g * (N_REF * 2) + r * 2;
    c0 += w * cp[0];
    c1 += w * cp[1];
  }
  if (wsum == 0.0f) wsum = 1.0f;
  float inv = 1.0f / wsum;
  out[idx * 2 + 0] = dist * (1.0f / 6.0f) + c0 * inv;   // P: 6.0 km/s
  out[idx * 2 + 1] = dist * (1.0f / 3.5f) + c1 * inv;   // S: 3.5 km/s
}

// ---------------------------------------------------------------------------
extern "C" void kernel_launch(void* const* d_in, const int* in_sizes, int n_in,
                              void* d_out, int out_size, void* d_ws, size_t ws_size,
                              hipStream_t stream) {
  const float* sta       = (const float*)d_in[0];
  const float* src       = (const float*)d_in[1];
  const float* x_grid    = (const float*)d_in[2];
  const float* locs_ref  = (const float*)d_in[3];
  const float* coefs     = (const float*)d_in[4];
  const float* coefs_ker = (const float*)d_in[5];
  float* out = (float*)d_out;

  int* sta_ind = (int*)d_ws;          // 200 ints
  int* nn_idx  = sta_ind + 256;       // 60000 ints (aligned past sta_ind)

  sta_ind_kernel<<<1, 256, 0, stream>>>(sta, locs_ref, sta_ind);

  // 250 waves of 16 sources, 8 waves per block
  int knn_blocks = (N_SRC / 16 + 7) / 8;
  knn_kernel<<<knn_blocks, 256, 0, stream>>>(src, x_grid, nn_idx);

  int pairs = N_SRC * N_STA;
  corr_kernel<<<(pairs + 255) / 256, 256, 0, stream>>>(
      sta, src, x_grid, coefs, coefs_ker, sta_ind, nn_idx, out);
}